// EnhancedGNN_39694087750251
// MI455X (gfx1250) — compile-verified
//
#include <hip/hip_runtime.h>
#include <math.h>

typedef float v2f __attribute__((ext_vector_type(2)));
typedef float v8f __attribute__((ext_vector_type(8)));

#define NEG_SLOPE 0.2f

// ---------------------------------------------------------------------------
// C[M x Nn] = A[M x K] * B[K x Nn], all row-major fp32, via V_WMMA_F32_16X16X4_F32.
// Block = 128 threads = 4 waves. Wave w owns columns [64w, 64w+64) (4 N-tiles).
// Each block processes TM row-tiles (TM*16 rows), so B fragments loaded once per
// k-step feed TM*4 WMMAs: 12 VMEM ops -> 16 WMMAs (TM=4).
// Tail: A-row loads clamped to M-1, stores for out-of-range tiles skipped
// (uniform per wave => EXEC all-ones around WMMA, as required).
// Requires Nn == 256, K % 4 == 0, M % 16 == 0.
// ---------------------------------------------------------------------------
template <int TM>
__global__ __launch_bounds__(128)
void gemm_f32_wmma_kernel(const float* __restrict__ A, const float* __restrict__ B,
                          float* __restrict__ C, int M, int K, int Nn) {
  const int lane = threadIdx.x & 31;
  const int wave = threadIdx.x >> 5;
  const int half = lane >> 4;   // 0 for lanes 0-15, 1 for lanes 16-31
  const int l16  = lane & 15;
  const int mt0  = blockIdx.x * TM;   // first row-tile index
  const int n0   = wave * 64;

  v8f acc[TM][4];
  #pragma unroll
  for (int t = 0; t < TM; ++t)
    #pragma unroll
    for (int j = 0; j < 4; ++j) acc[t][j] = (v8f){};

  // A fragment 16x4: lane<16 -> K = k+0,k+1 ; lane>=16 -> K = k+2,k+3
  const float* arow[TM];
  #pragma unroll
  for (int t = 0; t < TM; ++t) {
    int row = (mt0 + t) * 16 + l16;
    if (row >= M) row = M - 1;        // clamp; these tiles are never stored
    arow[t] = A + (size_t)row * K + half * 2;
  }

  for (int k = 0; k < K; k += 4) {
    // B fragment 4x16 per N-tile: lane indexes N, VGPR indexes K (mirrors A)
    const float* brow0 = B + (size_t)(k + half * 2) * Nn + n0 + l16;
    const float* brow1 = brow0 + Nn;
    v2f bf[4];
    #pragma unroll
    for (int j = 0; j < 4; ++j) bf[j] = (v2f){ brow0[j * 16], brow1[j * 16] };

    #pragma unroll
    for (int t = 0; t < TM; ++t) {
      v2f a = *(const v2f*)(arow[t] + k);
      #pragma unroll
      for (int j = 0; j < 4; ++j)
        acc[t][j] = __builtin_amdgcn_wmma_f32_16x16x4_f32(
            false, a, false, bf[j], (short)0, acc[t][j], false, false);
    }
  }

  // C/D layout: VGPR r: lanes 0-15 -> row r, lanes 16-31 -> row 8+r; lane -> column
  #pragma unroll
  for (int t = 0; t < TM; ++t) {
    const int mt = mt0 + t;
    if (mt * 16 >= M) break;          // uniform across the wave
    const int crow = mt * 16 + half * 8;
    #pragma unroll
    for (int r = 0; r < 8; ++r) {
      float* cp = C + (size_t)(crow + r) * Nn + n0 + l16;
      cp[0]  = acc[t][0][r];
      cp[16] = acc[t][1][r];
      cp[32] = acc[t][2][r];
      cp[48] = acc[t][3][r];
    }
  }
}

// ---------------------------------------------------------------------------
// es[n,h] = dot(h[n,h,:], a_src[h,:]) ; ed likewise. One wave per (n, head).
// ---------------------------------------------------------------------------
__global__ __launch_bounds__(256)
void attn_coef_kernel(const float* __restrict__ hmat, const float* __restrict__ a_src,
                      const float* __restrict__ a_dst, float* __restrict__ es,
                      float* __restrict__ ed, int N, int H, int D) {
  const int lane = threadIdx.x & 31;
  const long long pair = (long long)blockIdx.x * (blockDim.x >> 5) + (threadIdx.x >> 5);
  if (pair >= (long long)N * H) return;
  const int n  = (int)(pair / H);
  const int hh = (int)(pair % H);
  const float* hp = hmat + ((size_t)n * H + hh) * D;
  float s = 0.f, d = 0.f;
  for (int i = lane; i < D; i += 32) {
    float v = hp[i];
    s += v * a_src[hh * D + i];
    d += v * a_dst[hh * D + i];
  }
  #pragma unroll
  for (int o = 16; o > 0; o >>= 1) {
    s += __shfl_down(s, o, 32);
    d += __shfl_down(d, o, 32);
  }
  if (lane == 0) { es[pair] = s; ed[pair] = d; }
}

__global__ void fill_kernel(float* __restrict__ p, float v, long long n) {
  long long i = (long long)blockIdx.x * blockDim.x + threadIdx.x;
  if (i < n) p[i] = v;
}

// ---------------------------------------------------------------------------
// Pass 1: segment max of leaky_relu(es[src]+ed[dst]) over incoming edges of dst.
// Edges e in [0,E) come from edge_index; e in [E, E+N) are self loops (src=dst=e-E).
// ---------------------------------------------------------------------------
__global__ __launch_bounds__(256)
void edge_max_kernel(const float* __restrict__ es, const float* __restrict__ ed,
                     const int* __restrict__ src, const int* __restrict__ dst,
                     float* __restrict__ emax, long long Etot, long long E, int H) {
  long long t = (long long)blockIdx.x * blockDim.x + threadIdx.x;
  if (t >= Etot * H) return;
  long long e = t / H;
  int hh = (int)(t % H);
  int s = (e < E) ? src[e] : (int)(e - E);
  int d = (e < E) ? dst[e] : (int)(e - E);
  float v = es[(size_t)s * H + hh] + ed[(size_t)d * H + hh];
  v = (v > 0.f) ? v : NEG_SLOPE * v;
  __hip_atomic_fetch_max(&emax[(size_t)d * H + hh], v,
                         __ATOMIC_RELAXED, __HIP_MEMORY_SCOPE_AGENT);
}

// ---------------------------------------------------------------------------
// Pass 2 (fused): ex = exp(e - emax[dst]); denom[dst,h] += ex;
//                 acc[dst,h,:] += ex * hmat[src,h,:].
// One wave per (edge, head): lanes stripe D (coalesced 128B gathers of h[src],
// native global f32 atomic adds on acc[dst]).
// ---------------------------------------------------------------------------
__global__ __launch_bounds__(256)
void edge_accum_kernel(const float* __restrict__ hmat, const float* __restrict__ es,
                       const float* __restrict__ ed, const float* __restrict__ emax,
                       const int* __restrict__ src, const int* __restrict__ dst,
                       float* __restrict__ denom, float* __restrict__ acc,
                       long long Etot, long long E, int H, int D) {
  const int lane = threadIdx.x & 31;
  long long wid = (long long)blockIdx.x * (blockDim.x >> 5) + (threadIdx.x >> 5);
  if (wid >= Etot * H) return;
  long long e = wid / H;
  int hh = (int)(wid % H);
  int s = (e < E) ? src[e] : (int)(e - E);
  int d = (e < E) ? dst[e] : (int)(e - E);
  float v = es[(size_t)s * H + hh] + ed[(size_t)d * H + hh];
  v = (v > 0.f) ? v : NEG_SLOPE * v;
  float ex = expf(v - emax[(size_t)d * H + hh]);
  if (lane == 0) atomicAdd(&denom[(size_t)d * H + hh], ex);
  const float* hp = hmat + ((size_t)s * H + hh) * D;
  float*       ap = acc  + ((size_t)d * H + hh) * D;
  for (int i = lane; i < D; i += 32) atomicAdd(&ap[i], ex * hp[i]);
}

// ---------------------------------------------------------------------------
// out[n, h*D+d] = elu(acc[n,h,d] / denom[n,h] + b[h*D+d]), in place on acc.
// (Layer 2: H=1 so concat/mean are both identity.)
// ---------------------------------------------------------------------------
__global__ __launch_bounds__(256)
void finalize_kernel(float* __restrict__ acc, const float* __restrict__ denom,
                     const float* __restrict__ b, long long N, int H, int D) {
  long long i = (long long)blockIdx.x * blockDim.x + threadIdx.x;
  long long total = N * H * D;
  if (i >= total) return;
  int f  = (int)(i % (H * D));
  int hh = f / D;
  long long n = i / (H * D);
  float v = acc[i] / denom[n * H + hh] + b[f];
  acc[i] = (v > 0.f) ? v : (expf(v) - 1.0f);
}

static inline int cdiv(long long a, long long b) { return (int)((a + b - 1) / b); }

extern "C" void kernel_launch(void* const* d_in, const int* in_sizes, int n_in,
                              void* d_out, int out_size, void* d_ws, size_t ws_size,
                              hipStream_t stream) {
  const float* x      = (const float*)d_in[0];
  const int*   ei     = (const int*)d_in[1];
  const float* W1     = (const float*)d_in[2];
  const float* a_src1 = (const float*)d_in[3];
  const float* a_dst1 = (const float*)d_in[4];
  const float* b1     = (const float*)d_in[5];
  const float* W2     = (const float*)d_in[6];
  const float* a_src2 = (const float*)d_in[7];
  const float* a_dst2 = (const float*)d_in[8];
  const float* b2     = (const float*)d_in[9];

  const int F = 256;                       // F_in = H1*D1 = F_out = 256
  const int N = in_sizes[0] / F;
  const long long E    = in_sizes[1] / 2;  // edge_index is [2, E]
  const long long Etot = E + N;            // + self loops
  const int H1 = 8, D1 = 32;
  const int H2 = 1, D2 = 256;

  const int* srcp = ei;
  const int* dstp = ei + E;

  // Workspace layout
  char*  ws    = (char*)d_ws;
  float* hbuf  = (float*)ws;                                       // N*F
  float* acc   = (float*)(ws + (size_t)N * F * sizeof(float));     // N*F (acc -> x1)
  float* es    = (float*)(ws + (size_t)2 * N * F * sizeof(float)); // N*H1
  float* ed    = es    + (size_t)N * H1;
  float* emax  = ed    + (size_t)N * H1;
  float* denom = emax  + (size_t)N * H1;
  float* outp  = (float*)d_out;                                    // N*F

  const int TM = 4;
  const int gemm_blocks = cdiv(N / 16, TM);

  // ---------------- Layer 1 (H=8, D=32, concat) ----------------
  gemm_f32_wmma_kernel<4><<<gemm_blocks, 128, 0, stream>>>(x, W1, hbuf, N, F, F);
  attn_coef_kernel<<<cdiv((long long)N * H1, 8), 256, 0, stream>>>(
      hbuf, a_src1, a_dst1, es, ed, N, H1, D1);
  fill_kernel<<<cdiv((long long)N * H1, 256), 256, 0, stream>>>(emax, -INFINITY, (long long)N * H1);
  fill_kernel<<<cdiv((long long)N * H1, 256), 256, 0, stream>>>(denom, 0.f, (long long)N * H1);
  fill_kernel<<<cdiv((long long)N * F, 256), 256, 0, stream>>>(acc, 0.f, (long long)N * F);
  edge_max_kernel<<<cdiv(Etot * H1, 256), 256, 0, stream>>>(
      es, ed, srcp, dstp, emax, Etot, E, H1);
  edge_accum_kernel<<<cdiv(Etot * H1, 8), 256, 0, stream>>>(
      hbuf, es, ed, emax, srcp, dstp, denom, acc, Etot, E, H1, D1);
  finalize_kernel<<<cdiv((long long)N * F, 256), 256, 0, stream>>>(
      acc, denom, b1, N, H1, D1);                                  // acc now holds x1

  // ---------------- Layer 2 (H=1, D=256, mean==identity) ----------------
  gemm_f32_wmma_kernel<4><<<gemm_blocks, 128, 0, stream>>>(acc, W2, hbuf, N, F, F);
  attn_coef_kernel<<<cdiv((long long)N * H2, 8), 256, 0, stream>>>(
      hbuf, a_src2, a_dst2, es, ed, N, H2, D2);
  fill_kernel<<<cdiv((long long)N * H2, 256), 256, 0, stream>>>(emax, -INFINITY, (long long)N * H2);
  fill_kernel<<<cdiv((long long)N * H2, 256), 256, 0, stream>>>(denom, 0.f, (long long)N * H2);
  fill_kernel<<<cdiv((long long)N * F, 256), 256, 0, stream>>>(outp, 0.f, (long long)N * F);
  edge_max_kernel<<<cdiv(Etot * H2, 256), 256, 0, stream>>>(
      es, ed, srcp, dstp, emax, Etot, E, H2);
  edge_accum_kernel<<<cdiv(Etot * H2, 8), 256, 0, stream>>>(
      hbuf, es, ed, emax, srcp, dstp, denom, outp, Etot, E, H2, D2);
  finalize_kernel<<<cdiv((long long)N * F, 256), 256, 0, stream>>>(
      outp, denom, b2, N, H2, D2);
}